// FinancialEntityGraph_39556648796598
// MI455X (gfx1250) — compile-verified
//
#include <hip/hip_runtime.h>
#include <hip/hip_bf16.h>

#define N_ENT 1024
#define HDIM  128
#define HEADS 8
#define CHEAD 16

typedef __attribute__((ext_vector_type(2))) float v2f;
typedef __attribute__((ext_vector_type(8))) float v8f;

// D = A(16x4 f32) * B(4x16 f32) + C(16x16 f32), full-precision WMMA.
__device__ __forceinline__ v8f wmma_k4(v2f a, v2f b, v8f c) {
  return __builtin_amdgcn_wmma_f32_16x16x4_f32(false, a, false, b, (short)0, c,
                                               false, false);
}

// ---------------------------------------------------------------------------
// Out[1024x128] = X[1024x128] @ W[128x128] (+bias) (+addend), one wave per
// 16x16 tile, K unrolled into 32 v_wmma_f32_16x16x4_f32 ops.
// ---------------------------------------------------------------------------
__global__ __launch_bounds__(32)
void gemm_xw_kernel(const float* __restrict__ X, const float* __restrict__ W,
                    const float* __restrict__ bias,
                    const float* __restrict__ addend, float* __restrict__ Out) {
  const int it = blockIdx.x;   // row tile (0..63)
  const int jt = blockIdx.y;   // col tile (0..7)
  const int lane = threadIdx.x;
  const int hi = lane >> 4;
  const int lm = lane & 15;

  const float* xrow = X + (it * 16 + lm) * HDIM;  // A: lane holds row M=lm
  const float* wcol = W + jt * 16 + lm;           // B: lane holds col N=lm

  v8f acc = {0.f, 0.f, 0.f, 0.f, 0.f, 0.f, 0.f, 0.f};
#pragma unroll
  for (int kk = 0; kk < HDIM; kk += 4) {
    const int ka = kk + 2 * hi;  // lanes 16-31 carry K pair {2,3}
    v2f a; a.x = xrow[ka];            a.y = xrow[ka + 1];
    v2f b; b.x = wcol[ka * HDIM];     b.y = wcol[(ka + 1) * HDIM];
    acc = wmma_k4(a, b, acc);
  }

  const int col = jt * 16 + lm;
  const float bv = bias ? bias[col] : 0.0f;
#pragma unroll
  for (int r = 0; r < 8; ++r) {
    const int row = it * 16 + r + 8 * hi;  // C/D layout: vgpr r -> M = r + 8*hi
    float o = acc[r] + bv;
    if (addend) o += addend[row * HDIM + col];
    Out[row * HDIM + col] = o;
  }
}

// ---------------------------------------------------------------------------
// Pairwise edge predictor, written TRANSPOSED: wT[target][source],
// mT[target][source].  logit(s,t) = relu(aE[s]+bE[t]) . W2 + b2
// (b1 already folded into bE by the GEMM stage).
// ---------------------------------------------------------------------------
__global__ __launch_bounds__(256)
void edge_kernel(const float* __restrict__ aE, const float* __restrict__ bE,
                 const float* __restrict__ W2, const float* __restrict__ b2,
                 float* __restrict__ wT, float* __restrict__ mT) {
  __shared__ float sa[16][HDIM + 1];  // source rows (pad: conflict-free)
  __shared__ float sb[16][HDIM + 1];  // target rows
  __shared__ float sw2[HDIM];

  const int ts = blockIdx.x * 16;  // source base (column of wT)
  const int tt = blockIdx.y * 16;  // target base (row of wT)
  const int tid = threadIdx.y * 16 + threadIdx.x;

  for (int idx = tid; idx < 16 * HDIM; idx += 256) {
    const int r = idx >> 7, c = idx & (HDIM - 1);
    sa[r][c] = aE[(ts + r) * HDIM + c];
    sb[r][c] = bE[(tt + r) * HDIM + c];
  }
  if (tid < HDIM) sw2[tid] = W2[tid];
  __syncthreads();

  const int sj = threadIdx.x;  // source within tile
  const int ti = threadIdx.y;  // target within tile
  float acc = 0.f;
#pragma unroll 8
  for (int hh = 0; hh < HDIM; ++hh) {
    const float s = sa[sj][hh] + sb[ti][hh];
    acc = fmaf(fmaxf(s, 0.f), sw2[hh], acc);
  }
  const float logit = acc + b2[0];
  const int row = tt + ti, col = ts + sj;
  wT[row * N_ENT + col] = 1.f / (1.f + __expf(-logit));
  mT[row * N_ENT + col] = (logit > 0.f) ? 1.f : 0.f;
}

// ---------------------------------------------------------------------------
// Fused flash-style TransformerConv head block: one wave per (row-tile, head).
// S = Q K^T (WMMA) + edge term, online softmax in LDS, O += P V (WMMA),
// epilogue adds aw/L * We and the residual+skip base.
// ---------------------------------------------------------------------------
__global__ __launch_bounds__(32)
void attn_kernel(const float* __restrict__ q, const float* __restrict__ k,
                 const float* __restrict__ v, const float* __restrict__ wT,
                 const float* __restrict__ mT, const float* __restrict__ We,
                 const float* __restrict__ base, float* __restrict__ xout) {
  const int it = blockIdx.x;  // 0..63 row tile
  const int h  = blockIdx.y;  // head
  const int lane = threadIdx.x;
  const int hi = lane >> 4;
  const int lm = lane & 15;

  __shared__ float pbuf[16][17];
  __shared__ float wbuf[16][17];
  __shared__ float qwe[16];
  __shared__ float sM[16], sL[16], sAW[16], sCorr[16];

  const float* qrow = q + (it * 16 + lm) * HDIM + h * CHEAD;
  v2f qa[4];
#pragma unroll
  for (int s = 0; s < 4; ++s) {
    const int ka = s * 4 + 2 * hi;
    qa[s].x = qrow[ka];
    qa[s].y = qrow[ka + 1];
  }
  if (lane < 16) {
    float sdot = 0.f;
#pragma unroll
    for (int c = 0; c < CHEAD; ++c) sdot += qrow[c] * We[h * CHEAD + c];
    qwe[lane] = sdot;
    sM[lane] = -3.0e38f;
    sL[lane] = 0.f;
    sAW[lane] = 0.f;
  }
  __syncthreads();

  v8f o = {0.f, 0.f, 0.f, 0.f, 0.f, 0.f, 0.f, 0.f};
  const float scale = 0.25f;  // 1/sqrt(CHEAD)

  for (int jt2 = 0; jt2 < N_ENT / 16; ++jt2) {
    // ---- S = Q K^T ----
    v8f sacc = {0.f, 0.f, 0.f, 0.f, 0.f, 0.f, 0.f, 0.f};
    const float* krow = k + (jt2 * 16 + lm) * HDIM + h * CHEAD;
#pragma unroll
    for (int s = 0; s < 4; ++s) {
      const int ka = s * 4 + 2 * hi;
      v2f b; b.x = krow[ka]; b.y = krow[ka + 1];
      sacc = wmma_k4(qa[s], b, sacc);
    }
    // ---- logits + mask, staged to LDS ----
    const int jj = jt2 * 16 + lm;
#pragma unroll
    for (int r = 0; r < 8; ++r) {
      const int row = r + 8 * hi;
      const float wv = wT[(it * 16 + row) * N_ENT + jj];
      const float mv = mT[(it * 16 + row) * N_ENT + jj];
      float lg = (sacc[r] + qwe[row] * wv) * scale;
      lg = (mv != 0.f) ? lg : -1.0e9f;  // NEG: masked exp underflows to 0
      pbuf[row][lm] = lg;
      wbuf[row][lm] = wv;
    }
    __syncthreads();
    // ---- online softmax bookkeeping (16 lanes, one row each) ----
    if (lane < 16) {
      const int row = lane;
      float rmax = -3.0e38f;
#pragma unroll
      for (int c = 0; c < 16; ++c) rmax = fmaxf(rmax, pbuf[row][c]);
      const float newM = fmaxf(sM[row], rmax);
      const float corr = __expf(sM[row] - newM);
      float rsum = 0.f, rw = 0.f;
#pragma unroll
      for (int c = 0; c < 16; ++c) {
        const float e = __expf(pbuf[row][c] - newM);
        pbuf[row][c] = e;  // P in place
        rsum += e;
        rw += e * wbuf[row][c];
      }
      sM[row] = newM;
      sL[row] = sL[row] * corr + rsum;
      sAW[row] = sAW[row] * corr + rw;
      sCorr[row] = corr;
    }
    __syncthreads();
    // ---- rescale O, then O += P V ----
#pragma unroll
    for (int r = 0; r < 8; ++r) o[r] *= sCorr[r + 8 * hi];
    const float* vcol = v + h * CHEAD + lm;  // B: n = c = lm
#pragma unroll
    for (int s = 0; s < 4; ++s) {
      const int ka = s * 4 + 2 * hi;  // k = j within tile
      v2f a; a.x = pbuf[lm][ka]; a.y = pbuf[lm][ka + 1];
      v2f b; b.x = vcol[(jt2 * 16 + ka) * HDIM];
             b.y = vcol[(jt2 * 16 + ka + 1) * HDIM];
      o = wmma_k4(a, b, o);
    }
    __syncthreads();  // protect pbuf/wbuf before next tile rewrites them
  }

  // ---- epilogue: /L, + aw/L * We, + (x + x@Ws + bs) ----
#pragma unroll
  for (int r = 0; r < 8; ++r) {
    const int row = r + 8 * hi;
    const int gi = it * 16 + row;
    float attn = 0.f;
    if (sM[row] > -0.9e9f) {  // row has at least one unmasked edge
      const float inv = 1.f / sL[row];
      attn = o[r] * inv + (sAW[row] * inv) * We[h * CHEAD + lm];
    }
    const int idx = gi * HDIM + h * CHEAD + lm;
    xout[idx] = base[idx] + attn;
  }
}

// ---------------------------------------------------------------------------
extern "C" void kernel_launch(void* const* d_in, const int* in_sizes, int n_in,
                              void* d_out, int out_size, void* d_ws,
                              size_t ws_size, hipStream_t stream) {
  (void)in_sizes; (void)n_in; (void)out_size; (void)ws_size;
  // inputs per setup_inputs() order; mention_* are unused by the reference.
  const float* x0 = (const float*)d_in[2];   // entity_embeddings [1024,128]
  const float* W1 = (const float*)d_in[3];   // [256,128]
  const float* b1 = (const float*)d_in[4];   // [128]
  const float* W2 = (const float*)d_in[5];   // [128,1]
  const float* b2 = (const float*)d_in[6];   // [1]
  const float* Wq = (const float*)d_in[7];   // [3,128,128]
  const float* bq = (const float*)d_in[8];   // [3,128]
  const float* Wk = (const float*)d_in[9];
  const float* bk = (const float*)d_in[10];
  const float* Wv = (const float*)d_in[11];
  const float* bv = (const float*)d_in[12];
  const float* We = (const float*)d_in[13];  // [3,128]
  const float* Wsk = (const float*)d_in[14]; // [3,128,128]
  const float* bsk = (const float*)d_in[15]; // [3,128]

  float* ws = (float*)d_ws;
  float* wT = ws;                          // [1024,1024]
  float* mT = wT + N_ENT * N_ENT;          // [1024,1024]
  float* aE = mT + N_ENT * N_ENT;          // [1024,128]
  float* bE = aE + N_ENT * HDIM;
  float* qb = bE + N_ENT * HDIM;
  float* kb = qb + N_ENT * HDIM;
  float* vb = kb + N_ENT * HDIM;
  float* baseb = vb + N_ENT * HDIM;
  float* xA = baseb + N_ENT * HDIM;
  float* xB = xA + N_ENT * HDIM;

  const dim3 gG(N_ENT / 16, HDIM / 16);  // 64 x 8 WMMA tiles
  const dim3 bW(32);                     // one wave per block

  // Edge graph built once from the initial embeddings (b1 folded into bE).
  gemm_xw_kernel<<<gG, bW, 0, stream>>>(x0, W1, nullptr, nullptr, aE);
  gemm_xw_kernel<<<gG, bW, 0, stream>>>(x0, W1 + HDIM * HDIM, b1, nullptr, bE);
  edge_kernel<<<dim3(N_ENT / 16, N_ENT / 16), dim3(16, 16), 0, stream>>>(
      aE, bE, W2, b2, wT, mT);

  const float* xin = x0;
  for (int l = 0; l < 3; ++l) {
    const float* Wq_l = Wq + l * HDIM * HDIM;
    const float* Wk_l = Wk + l * HDIM * HDIM;
    const float* Wv_l = Wv + l * HDIM * HDIM;
    const float* Ws_l = Wsk + l * HDIM * HDIM;
    float* xout = (l == 2) ? (float*)d_out : ((l == 0) ? xA : xB);

    gemm_xw_kernel<<<gG, bW, 0, stream>>>(xin, Wq_l, bq + l * HDIM, nullptr, qb);
    gemm_xw_kernel<<<gG, bW, 0, stream>>>(xin, Wk_l, bk + l * HDIM, nullptr, kb);
    gemm_xw_kernel<<<gG, bW, 0, stream>>>(xin, Wv_l, bv + l * HDIM, nullptr, vb);
    // base = x + x @ Ws + bs  (residual + skip, fused)
    gemm_xw_kernel<<<gG, bW, 0, stream>>>(xin, Ws_l, bsk + l * HDIM, xin, baseb);

    attn_kernel<<<dim3(N_ENT / 16, HEADS), bW, 0, stream>>>(
        qb, kb, vb, wT, mT, We + l * HDIM, baseb, xout);
    xin = xout;
  }
}